// CGMMLayer_0_9732395893090
// MI455X (gfx1250) — compile-verified
//
#include <hip/hip_runtime.h>

typedef __attribute__((ext_vector_type(2))) float v2f;
typedef __attribute__((ext_vector_type(8))) float v8f;

#define CC 10
#define MM 50
#define NGEN 8
#define TABSZ (MM * NGEN)          // 400
#define LIK_STRIDE 16              // padded stride: cols 8..15 are zero
#define LIKSZ (MM * LIK_STRIDE)    // 800

#define WAVE_TILE 256              // nodes per wave
#define WAVES_PER_BLOCK 8          // 256 threads, wave32
#define BLOCK_TILE (WAVE_TILE * WAVES_PER_BLOCK)

// ---------------------------------------------------------------------------
// Kernel 1: precompute per-label tables.
//   lik_tab[m][g] = sum_c posterior[c]*log(numerator[c])
//   h_tab[m][g]   = (float) argmax_c posterior[c]   (first max, like jnp.argmax)
// 400 (m,g) pairs, one thread each. Tiny.
// ---------------------------------------------------------------------------
__global__ void cgmm_build_tables(const float* __restrict__ B,
                                  const float* __restrict__ Pi,
                                  float* __restrict__ lik_tab,
                                  float* __restrict__ h_tab) {
    int t = threadIdx.x;
    if (t >= TABSZ) return;
    int m = t / NGEN;
    int g = t % NGEN;

    // softmax over C for Pi[:, g]
    float pimax = -1e30f;
    for (int c = 0; c < CC; ++c) pimax = fmaxf(pimax, Pi[c * NGEN + g]);
    float pie[CC];
    float pisum = 0.0f;
    for (int c = 0; c < CC; ++c) {
        pie[c] = expf(Pi[c * NGEN + g] - pimax);
        pisum += pie[c];
    }

    // numerator[c] = softmax_M(B[c,:,g])[m] * softmax_C(Pi[:,g])[c]
    float num[CC];
    float s = 0.0f;
    for (int c = 0; c < CC; ++c) {
        const float* Bc = B + (size_t)c * MM * NGEN + g;   // stride NGEN over m'
        float bmax = -1e30f;
        for (int mp = 0; mp < MM; ++mp) bmax = fmaxf(bmax, Bc[mp * NGEN]);
        float bsum = 0.0f;
        for (int mp = 0; mp < MM; ++mp) bsum += expf(Bc[mp * NGEN] - bmax);
        float smB  = expf(Bc[m * NGEN] - bmax) / bsum;
        num[c] = (pie[c] / pisum) * smB;
        s += num[c];
    }

    float lik  = 0.0f;
    float best = num[0];
    int   amax = 0;
    for (int c = 0; c < CC; ++c) {
        lik += (num[c] / s) * logf(num[c]);
        if (num[c] > best) { best = num[c]; amax = c; }
    }
    lik_tab[t] = lik;
    h_tab[t]   = (float)amax;
}

// ---------------------------------------------------------------------------
// Kernel 2: zero the likelihood output region (harness poisons d_out).
// ---------------------------------------------------------------------------
__global__ void cgmm_zero(float* __restrict__ p, int n) {
    int i = blockIdx.x * blockDim.x + threadIdx.x;
    if (i < n) p[i] = 0.0f;
}

// ---------------------------------------------------------------------------
// Kernel 3: stream nodes. Per wave: stage 256 nodes of x/batch into LDS,
// write h_states (float4 pairs, coalesced), then segment-sum likelihood via
// V_WMMA_F32_16X16X4_F32 (A = 16x4 segment indicator, B = 4x16 lik rows),
// flushing one f32 atomic per touched (segment, gen) instead of per node.
// The likelihood table is staged at stride 16 with zero-padded cols 8..15 so
// the WMMA inner loop is completely branch-free (EXEC stays all-ones).
// Fallback (wave-uniform) scalar-atomic path if the tile spans >=16 graphs.
// ---------------------------------------------------------------------------
__global__ void __launch_bounds__(WAVES_PER_BLOCK * 32)
cgmm_main(const int* __restrict__ x, const int* __restrict__ batch,
          const float* __restrict__ lik_tab, const float* __restrict__ h_tab,
          float* __restrict__ out_lik, float* __restrict__ out_h, int n_nodes) {
    __shared__ float s_lik16[LIKSZ];   // [50][16], cols 8..15 = 0
    __shared__ float s_h[TABSZ];       // [50][8]
    __shared__ int   s_x[BLOCK_TILE];
    __shared__ int   s_seg[BLOCK_TILE];

    const int tid = threadIdx.x;

    // Stage tables once per block (lik padded to stride 16 with zeros).
    for (int i = tid; i < LIKSZ; i += blockDim.x) {
        int m = i >> 4, col = i & 15;
        s_lik16[i] = (col < NGEN) ? lik_tab[m * NGEN + col] : 0.0f;
    }
    for (int i = tid; i < TABSZ; i += blockDim.x) {
        s_h[i] = h_tab[i];
    }
    __syncthreads();

    const int wave  = tid >> 5;          // gfx1250 is wave32-only
    const int lane  = tid & 31;
    const int wofs  = wave * WAVE_TILE;  // this wave's LDS window
    const int nbase = blockIdx.x * BLOCK_TILE + wofs;

    // Stage x/batch for this wave's 256 nodes; emit h_states on the fly.
    for (int j = 0; j < WAVE_TILE / 32; ++j) {
        int n  = nbase + j * 32 + lane;
        int xm = 0, sg = -1;
        if (n < n_nodes) {
            xm = x[n];
            sg = batch[n];
            const float4* ht = (const float4*)&s_h[xm * NGEN]; // 32B aligned
            float4* hp = (float4*)(out_h + (size_t)n * NGEN);
            hp[0] = ht[0];
            hp[1] = ht[1];
        }
        s_x[wofs + j * 32 + lane]   = xm;
        s_seg[wofs + j * 32 + lane] = sg;
    }
    __syncthreads();

    if (nbase >= n_nodes) return;        // wave-uniform; after last barrier

    const int count = min(WAVE_TILE, n_nodes - nbase);
    const int first = s_seg[wofs];                 // wave-uniform
    const int span  = s_seg[wofs + count - 1] - first;

    if (span < 16) {
        // WMMA segment-sum. batch is sorted, so the tile's graphs map to
        // local segments 0..span (< 16). Invalid nodes carry seg = -1 and
        // contribute a zero indicator column.
        v8f acc = {};
        const int ka0    = (lane >> 4) * 2;        // K indices owned by lane
        const int ka1    = ka0 + 1;
        const int rowcol = lane & 15;              // A row / B col / D col
        for (int k = 0; k < WAVE_TILE; k += 4) {
            v2f a, b;
            int sg0 = s_seg[wofs + k + ka0];       // pair -> ds_load_b64
            int sg1 = s_seg[wofs + k + ka1];
            a.x = (sg0 - first == rowcol) ? 1.0f : 0.0f;
            a.y = (sg1 - first == rowcol) ? 1.0f : 0.0f;
            int x0 = s_x[wofs + k + ka0];          // pair -> ds_load_b64
            int x1 = s_x[wofs + k + ka1];
            b.x = s_lik16[x0 * LIK_STRIDE + rowcol];   // branch-free (padded)
            b.y = s_lik16[x1 * LIK_STRIDE + rowcol];
            acc = __builtin_amdgcn_wmma_f32_16x16x4_f32(
                false, a, false, b, (short)0, acc, false, false);
        }
        // D layout: lane l, VGPR v -> [M = v + (l<16?0:8), N = l&15].
        if (rowcol < NGEN) {
            const int mofs = (lane < 16) ? 0 : 8;
#pragma unroll
            for (int v = 0; v < 8; ++v) {
                int seg = v + mofs;
                if (seg <= span) {
                    atomicAdd(&out_lik[(size_t)(first + seg) * NGEN + rowcol],
                              acc[v]);
                }
            }
        }
    } else {
        // Rare: tile spans >=16 graphs. Wave-uniform scalar-atomic fallback.
        for (int j = 0; j < WAVE_TILE / 32; ++j) {
            int idx = wofs + j * 32 + lane;
            int sg  = s_seg[idx];
            if (sg >= 0) {
                int xm = s_x[idx];
#pragma unroll
                for (int g = 0; g < NGEN; ++g) {
                    atomicAdd(&out_lik[(size_t)sg * NGEN + g],
                              s_lik16[xm * LIK_STRIDE + g]);
                }
            }
        }
    }
}

// ---------------------------------------------------------------------------
// Launch
//   d_in: x [N] int32, batch [N] int32, B [10*50*8] f32, Pi [10*8] f32
//   d_out: likelihood [G*8] f32  ++  h_states [N*8] (as f32 indices)
// ---------------------------------------------------------------------------
extern "C" void kernel_launch(void* const* d_in, const int* in_sizes, int n_in,
                              void* d_out, int out_size, void* d_ws, size_t ws_size,
                              hipStream_t stream) {
    const int*   x     = (const int*)d_in[0];
    const int*   batch = (const int*)d_in[1];
    const float* B     = (const float*)d_in[2];
    const float* Pi    = (const float*)d_in[3];

    const int n_nodes  = in_sizes[0];
    const int n_graphs = out_size / NGEN - n_nodes;   // out = G*8 + N*8

    float* out_lik = (float*)d_out;
    float* out_h   = out_lik + (size_t)n_graphs * NGEN;

    float* lik_tab = (float*)d_ws;
    float* h_tab   = lik_tab + TABSZ;

    cgmm_build_tables<<<1, 512, 0, stream>>>(B, Pi, lik_tab, h_tab);

    int zl = n_graphs * NGEN;
    cgmm_zero<<<(zl + 255) / 256, 256, 0, stream>>>(out_lik, zl);

    int nblocks = (n_nodes + BLOCK_TILE - 1) / BLOCK_TILE;
    cgmm_main<<<nblocks, WAVES_PER_BLOCK * 32, 0, stream>>>(
        x, batch, lik_tab, h_tab, out_lik, out_h, n_nodes);
}